// triangulation_layer_36532991820483
// MI455X (gfx1250) — compile-verified
//
#include <hip/hip_runtime.h>

typedef __attribute__((ext_vector_type(2))) float v2f;
typedef __attribute__((ext_vector_type(8))) float v8f;

#define TEMP_F 10000.0f

// One block per (map, b, p) slice: online softmax expectation over 256x256.
// 256 threads (8 wave32), each streams 64 float4 (b128) loads.
__global__ __launch_bounds__(256) void softargmax_kernel(
    const float* __restrict__ ap, const float* __restrict__ lat,
    float* __restrict__ uw)
{
  const int slice = blockIdx.x;   // b*64 + p, 0..511
  const int map   = blockIdx.y;   // 0 = ap, 1 = lat
  const float*  src  = (map == 0 ? ap : lat) + (size_t)slice * 65536u;
  const float4* src4 = (const float4*)src;
  const int t = threadIdx.x;

  float m = -3.402823466e38f, l = 0.f, sx = 0.f, sy = 0.f;

  #pragma unroll 4
  for (int it = 0; it < 64; ++it) {
    const int idx4 = it * 256 + t;
    __builtin_prefetch(&src4[idx4 + 8 * 256], 0, 0);  // global_prefetch_b8, ~8 iters ahead
    float4 v = src4[idx4];
    const int f  = idx4 << 2;            // flat element index (W=256)
    const float y  = (float)(f >> 8);    // row
    const float x0 = (float)(f & 255);   // first column of the 4
    float t0 = TEMP_F * v.x, t1 = TEMP_F * v.y,
          t2 = TEMP_F * v.z, t3 = TEMP_F * v.w;
    float vm = fmaxf(fmaxf(t0, t1), fmaxf(t2, t3));
    if (vm > m) {                        // rare: rescale running sums
      float s = __expf(m - vm);
      l *= s; sx *= s; sy *= s; m = vm;
    }
    float p0 = __expf(t0 - m), p1 = __expf(t1 - m),
          p2 = __expf(t2 - m), p3 = __expf(t3 - m);
    l += (p0 + p1) + (p2 + p3);
    sx = fmaf(p0, x0,        sx); sx = fmaf(p1, x0 + 1.f, sx);
    sx = fmaf(p2, x0 + 2.f,  sx); sx = fmaf(p3, x0 + 3.f, sx);
    sy = fmaf((p0 + p1) + (p2 + p3), y, sy);
  }

  __shared__ float sdata[768];

  // Block max.
  sdata[t] = m;
  __syncthreads();
  for (int s = 128; s > 0; s >>= 1) {
    if (t < s) sdata[t] = fmaxf(sdata[t], sdata[t + s]);
    __syncthreads();
  }
  const float M = sdata[0];
  __syncthreads();

  // Rescale partials to the common max and stage for the WMMA reduction.
  const float sc = __expf(m - M);
  sdata[t]       = l  * sc;   // q=0: sum of weights
  sdata[256 + t] = sx * sc;   // q=1: sum of w*x
  sdata[512 + t] = sy * sc;   // q=2: sum of w*y
  __syncthreads();

  // Wave 0 (EXEC all-ones: wave-uniform branch) sums each 256-vector with
  // chained V_WMMA_F32_16X16X4_F32 against an all-ones B (exact f32 row-sums).
  if (t < 32) {
    const int lane = t;
    const int row  = lane & 15;   // A-matrix M index
    const int hi   = lane >> 4;   // 0 -> K0/K1, 1 -> K2/K3 (ISA 16x4 A layout)
    v2f ones; ones.x = 1.0f; ones.y = 1.0f;
    float tot[3];
    #pragma unroll
    for (int q = 0; q < 3; ++q) {
      v8f c = {};
      #pragma unroll
      for (int ch = 0; ch < 4; ++ch) {   // 4 chunks of 64 values, C accumulates
        const float2* b2 = (const float2*)&sdata[q * 256 + ch * 64];
        float2 av = b2[2 * row + hi];    // A[row][2*hi], A[row][2*hi+1]
        v2f a; a.x = av.x; a.y = av.y;
        c = __builtin_amdgcn_wmma_f32_16x16x4_f32(false, a, false, ones,
                                                  (short)0, c, false, false);
      }
      // lanes 0-15 hold rowsums 0..7 in c[0..7]; lanes 16-31 hold rows 8..15
      float s = ((c[0] + c[1]) + (c[2] + c[3])) + ((c[4] + c[5]) + (c[6] + c[7]));
      tot[q] = __shfl(s, 0, 32) + __shfl(s, 16, 32);
    }
    if (lane == 0) {
      const float ex = tot[1] / tot[0];  // expected column
      const float ey = tot[2] / tot[0];  // expected row
      float* o = uw + ((size_t)map * 512 + slice) * 2;
      o[0] = ey - 128.0f;   // u = (y - W/2, x - H/2)
      o[1] = ex - 128.0f;
    }
  }
}

// 512 independent 4x3 least-squares solves: X = (A^T A)^-1 A^T b + 127.5
__global__ __launch_bounds__(256) void triangulate_kernel(
    const float* __restrict__ uw, float* __restrict__ out)
{
  const int idx = blockIdx.x * 256 + threadIdx.x;
  if (idx >= 512) return;
  const float u0 = uw[idx * 2 + 0],        u1 = uw[idx * 2 + 1];
  const float u2 = uw[1024 + idx * 2 + 0], u3 = uw[1024 + idx * 2 + 1];

  // inv(Tr_ap) / inv(Tr_lat) rotation+translation, computed analytically.
  const float RAP[3][3]  = {{1,0,0},{0,0,-1},{0,1,0}};
  const float TAP[3]     = {0,0,700};
  const float RLAT[3][3] = {{0,-1,0},{0,0,-1},{1,0,0}};
  const float TLAT[3]    = {0,0,700};
  float D1[2][3] = {{-1800.f, 0.f, u0}, {0.f, -1800.f, u1}};
  float D2[2][3] = {{-1800.f, 0.f, u2}, {0.f, -1800.f, u3}};

  float A[4][3], bb[4];
  for (int i = 0; i < 2; ++i) {
    for (int j = 0; j < 3; ++j) {
      float s1 = 0.f, s2 = 0.f;
      for (int k = 0; k < 3; ++k) {
        s1 += D1[i][k] * RAP[k][j];
        s2 += D2[i][k] * RLAT[k][j];
      }
      A[i][j] = s1; A[2 + i][j] = s2;
    }
    float dt1 = 0.f, dt2 = 0.f;
    for (int j = 0; j < 3; ++j) { dt1 += D1[i][j] * TAP[j]; dt2 += D2[i][j] * TLAT[j]; }
    bb[i]     = -900.f * D1[i][2] - dt1;
    bb[2 + i] = -900.f * D2[i][2] - dt2;
  }

  double N[3][3] = {}, r[3] = {};
  for (int i = 0; i < 4; ++i)
    for (int j = 0; j < 3; ++j) {
      r[j] += (double)A[i][j] * (double)bb[i];
      for (int k = 0; k < 3; ++k) N[j][k] += (double)A[i][j] * (double)A[i][k];
    }
  const double C00 = N[1][1]*N[2][2] - N[1][2]*N[1][2];
  const double C01 = N[0][2]*N[1][2] - N[0][1]*N[2][2];
  const double C02 = N[0][1]*N[1][2] - N[0][2]*N[1][1];
  const double C11 = N[0][0]*N[2][2] - N[0][2]*N[0][2];
  const double C12 = N[0][1]*N[0][2] - N[0][0]*N[1][2];
  const double C22 = N[0][0]*N[1][1] - N[0][1]*N[0][1];
  const double det = N[0][0]*C00 + N[0][1]*C01 + N[0][2]*C02;
  const double inv = 1.0 / det;
  const double X0 = (C00*r[0] + C01*r[1] + C02*r[2]) * inv;
  const double X1 = (C01*r[0] + C11*r[1] + C12*r[2]) * inv;
  const double X2 = (C02*r[0] + C12*r[1] + C22*r[2]) * inv;
  out[idx * 3 + 0] = (float)(X0 + 127.5);
  out[idx * 3 + 1] = (float)(X1 + 127.5);
  out[idx * 3 + 2] = (float)(X2 + 127.5);
}

extern "C" void kernel_launch(void* const* d_in, const int* in_sizes, int n_in,
                              void* d_out, int out_size, void* d_ws, size_t ws_size,
                              hipStream_t stream) {
  const float* ap  = (const float*)d_in[0];   // (8,64,256,256) f32
  const float* lat = (const float*)d_in[1];   // (8,64,256,256) f32
  float* uw  = (float*)d_ws;                  // 2048 floats: [map][slice][2]
  float* out = (float*)d_out;                 // (8,64,3) f32

  softargmax_kernel<<<dim3(512, 2), 256, 0, stream>>>(ap, lat, uw);
  triangulate_kernel<<<dim3(2), 256, 0, stream>>>(uw, out);
}